// mynet_76295799046514
// MI455X (gfx1250) — compile-verified
//
#include <hip/hip_runtime.h>

#define LNEPS 1e-5f

typedef float v2f __attribute__((ext_vector_type(2)));
typedef float v8f __attribute__((ext_vector_type(8)));

__device__ __forceinline__ v8f wmma4(v2f a, v2f b, v8f c) {
  // D(16x16,f32) = A(16x4,f32) x B(4x16,f32) + C
  return __builtin_amdgcn_wmma_f32_16x16x4_f32(false, a, false, b, (short)0, c, false, false);
}

// ---------------------------------------------------------------- utilities
__global__ void zero_kernel(float* __restrict__ p, long long n) {
  long long i = (long long)blockIdx.x * blockDim.x + threadIdx.x;
  long long st = (long long)gridDim.x * blockDim.x;
  for (; i < n; i += st) p[i] = 0.f;
}

// ---------------------------------------------------------------- CNN branch
// conv1: in (32,1,1000,4) -> (32,32,951); 200-tap dot is contiguous in memory
__global__ void conv1_kernel(const float* __restrict__ of, const float* __restrict__ w1,
                             const float* __restrict__ b1, float* __restrict__ outp) {
  int t = blockIdx.x * blockDim.x + threadIdx.x;
  int bo = blockIdx.y;              // b*32 + oc
  if (t >= 951) return;
  int b = bo >> 5, oc = bo & 31;
  const float* ip = of + (size_t)b * 4000 + (size_t)t * 4;
  const float* wp = w1 + (size_t)oc * 200;
  float acc = b1[oc];
  for (int i = 0; i < 200; i++) acc += ip[i] * wp[i];
  outp[(size_t)bo * 951 + t] = acc;
}

// conv2 as implicit GEMM, M=oc(128), N=t(902), K=32*50=1600. + bias + bn + relu
__global__ void conv2_wmma(const float* __restrict__ im, const float* __restrict__ w2,
                           const float* __restrict__ b2, const float* __restrict__ bng,
                           const float* __restrict__ bnb, float* __restrict__ outp) {
  const int b = blockIdx.y;
  const int t0 = blockIdx.x * 16;
  const int lane = threadIdx.x & 31;
  const int wv = threadIdx.x >> 5;      // 8 waves -> 8 oc tiles
  const int oc0 = wv * 16;
  const int mrow = lane & 15;
  const int hf = lane >> 4;
  const int tn = t0 + mrow;
  const int tcl = (tn < 902) ? tn : 901;
  const float* imb = im + (size_t)b * 32 * 951;
  v8f c = {0.f,0.f,0.f,0.f,0.f,0.f,0.f,0.f};
  for (int k = 0; k < 1600; k += 4) {
    int kA = k + 2 * hf;
    v2f a = { w2[(size_t)(oc0 + mrow) * 1600 + kA],
              w2[(size_t)(oc0 + mrow) * 1600 + kA + 1] };
    int ic0 = kA / 50, r0 = kA % 50;
    int ic1 = (kA + 1) / 50, r1 = (kA + 1) % 50;
    v2f bb = { imb[(size_t)ic0 * 951 + tcl + r0],
               imb[(size_t)ic1 * 951 + tcl + r1] };
    c = wmma4(a, bb, c);
  }
  if (tn < 902) {
    for (int r = 0; r < 8; r++) {
      int oc = oc0 + r + 8 * hf;
      float sc = bng[oc] * rsqrtf(1.f + LNEPS);
      float v = (c[r] + b2[oc]) * sc + bnb[oc];
      outp[((size_t)b * 128 + oc) * 902 + tn] = fmaxf(v, 0.f);
    }
  }
}

// max over 902 positions per (b,oc) row; one wave per row
__global__ void maxred_kernel(const float* __restrict__ x, float* __restrict__ cm) {
  int row = blockIdx.x * 8 + (threadIdx.x >> 5);
  int lane = threadIdx.x & 31;
  const float* p = x + (size_t)row * 902;
  float m = -1e30f;
  for (int t = lane; t < 902; t += 32) m = fmaxf(m, p[t]);
  for (int o2 = 16; o2; o2 >>= 1) m = fmaxf(m, __shfl_xor(m, o2, 32));
  if (lane == 0) cm[row] = m;
}

// small dense fc: out(M,N) = X(M,K) @ W(K,N) + b
__global__ void fc_kernel(const float* __restrict__ X, const float* __restrict__ Wm,
                          const float* __restrict__ bias, float* __restrict__ outp,
                          int M, int K, int N) {
  int gid = blockIdx.x * blockDim.x + threadIdx.x;
  if (gid >= M * N) return;
  int mm = gid / N, j = gid % N;
  float a = bias[j];
  const float* xp = X + (size_t)mm * K;
  for (int k = 0; k < K; k++) a += xp[k] * Wm[(size_t)k * N + j];
  outp[gid] = a;
}

// ---------------------------------------------------------------- skinny (M=32) GEMM via WMMA f32
// out(32,N) = X(32,K) @ W(K,N) [+ bias]. mode=0: direct store + bias; mode=1: atomicAdd (K-split)
__global__ void gemm_skinny32(const float* __restrict__ X, const float* __restrict__ Wm,
                              const float* __restrict__ bias, float* __restrict__ outp,
                              int K, int N, int kchunk, int mode) {
  __shared__ float xs[32 * 64];
  const int lane = threadIdx.x & 31;
  const int wv = threadIdx.x >> 5;        // 4 waves -> 4 n-tiles (64 cols/block)
  const int n = blockIdx.x * 64 + wv * 16 + (lane & 15);
  const int mrow = lane & 15;
  const int hf = lane >> 4;
  const int kbeg = blockIdx.y * kchunk;
  const int kend = (kbeg + kchunk < K) ? (kbeg + kchunk) : K;
  v8f c0 = {0.f,0.f,0.f,0.f,0.f,0.f,0.f,0.f};
  v8f c1 = {0.f,0.f,0.f,0.f,0.f,0.f,0.f,0.f};
  for (int k0 = kbeg; k0 < kend; k0 += 64) {
    for (int i = threadIdx.x; i < 2048; i += 128) {
      int mm = i >> 6, kk = i & 63;
      int kg = k0 + kk;
      xs[mm * 64 + kk] = (kg < kend) ? X[(size_t)mm * K + kg] : 0.f;
    }
    __syncthreads();
    for (int kk = 0; kk < 64; kk += 4) {
      int kA = kk + 2 * hf;
      int kg = k0 + kA;
      v2f a0 = { xs[mrow * 64 + kA], xs[mrow * 64 + kA + 1] };
      v2f a1 = { xs[(mrow + 16) * 64 + kA], xs[(mrow + 16) * 64 + kA + 1] };
      float bv0 = (kg < kend) ? Wm[(size_t)kg * N + n] : 0.f;
      float bv1 = (kg + 1 < kend) ? Wm[(size_t)(kg + 1) * N + n] : 0.f;
      v2f b = { bv0, bv1 };
      c0 = wmma4(a0, b, c0);
      c1 = wmma4(a1, b, c1);
    }
    __syncthreads();
  }
  if (mode == 0) {
    float bb = bias ? bias[n] : 0.f;
    for (int r = 0; r < 8; r++) {
      int mr = r + 8 * hf;
      outp[(size_t)mr * N + n] = c0[r] + bb;
      outp[(size_t)(mr + 16) * N + n] = c1[r] + bb;
    }
  } else {
    for (int r = 0; r < 8; r++) {
      int mr = r + 8 * hf;
      atomicAdd(&outp[(size_t)mr * N + n], c0[r]);
      atomicAdd(&outp[(size_t)(mr + 16) * N + n], c1[r]);
    }
  }
}

// ---------------------------------------------------------------- graph aggregation
__global__ void count_kernel(const int* __restrict__ ei, int E, int E2, int offs,
                             int dstRow, float* __restrict__ cnt) {
  int e = blockIdx.x * blockDim.x + threadIdx.x;
  if (e >= E) return;
  int d = ei[(size_t)dstRow * E2 + 2 * e + offs];
  atomicAdd(&cnt[d], 1.f);
}

__global__ void agg3_kernel(const int* __restrict__ ei, int E, int E2, int offs,
                            int srcRow, int dstRow, const float* __restrict__ xsrc,
                            float* __restrict__ agg) {
  int e = blockIdx.x * blockDim.x + threadIdx.x;
  if (e >= E) return;
  int j = 2 * e + offs;
  int s = ei[(size_t)srcRow * E2 + j];
  int d = ei[(size_t)dstRow * E2 + j];
  for (int c = 0; c < 3; c++)
    atomicAdd(&agg[(size_t)d * 3 + c], xsrc[(size_t)s * 3 + c]);
}

// blockDim = 128 (one block per edge, lane = channel) -> coalesced gather + atomic
__global__ void agg128_kernel(const int* __restrict__ ei, int E2, int offs,
                              int srcRow, int dstRow, const float* __restrict__ xsrc,
                              float* __restrict__ agg) {
  int e = blockIdx.x;
  int c = threadIdx.x;
  int j = 2 * e + offs;
  int s = ei[(size_t)srcRow * E2 + j];
  int d = ei[(size_t)dstRow * E2 + j];
  atomicAdd(&agg[(size_t)d * 128 + c], xsrc[(size_t)s * 128 + c]);
}

// ---------------------------------------------------------------- fused SAGE layer
// out(M,128) = relu( (Sg/cnt) @ Wl + bl + Xo @ Wr ); grid.x = M/16, block=256 (8 n-tiles)
__global__ void sage_fused(const float* __restrict__ Sg, const float* __restrict__ cnt,
                           const float* __restrict__ Xo, const float* __restrict__ Wl,
                           const float* __restrict__ bl, const float* __restrict__ Wr,
                           float* __restrict__ outp, int Kl, int Kr) {
  __shared__ float sA[16 * 128];
  __shared__ float sX[16 * 128];
  const int m0 = blockIdx.x * 16;
  const int Klp = (Kl + 3) & ~3;
  const int Krp = (Kr + 3) & ~3;
  for (int i = threadIdx.x; i < 16 * Klp; i += 256) {
    int r = i / Klp, cix = i % Klp;
    sA[r * 128 + cix] = (cix < Kl) ? Sg[(size_t)(m0 + r) * Kl + cix] : 0.f;
  }
  for (int i = threadIdx.x; i < 16 * Krp; i += 256) {
    int r = i / Krp, cix = i % Krp;
    sX[r * 128 + cix] = (cix < Kr) ? Xo[(size_t)(m0 + r) * Kr + cix] : 0.f;
  }
  __syncthreads();
  const int lane = threadIdx.x & 31;
  const int wv = threadIdx.x >> 5;
  const int n = wv * 16 + (lane & 15);
  const int mrow = lane & 15;
  const int hf = lane >> 4;
  v8f cl = {0.f,0.f,0.f,0.f,0.f,0.f,0.f,0.f};
  v8f cr = {0.f,0.f,0.f,0.f,0.f,0.f,0.f,0.f};
  for (int k = 0; k < Klp; k += 4) {
    int kA = k + 2 * hf;
    v2f a = { sA[mrow * 128 + kA], sA[mrow * 128 + kA + 1] };
    v2f b = { (kA < Kl) ? Wl[(size_t)kA * 128 + n] : 0.f,
              (kA + 1 < Kl) ? Wl[(size_t)(kA + 1) * 128 + n] : 0.f };
    cl = wmma4(a, b, cl);
  }
  for (int k = 0; k < Krp; k += 4) {
    int kA = k + 2 * hf;
    v2f a = { sX[mrow * 128 + kA], sX[mrow * 128 + kA + 1] };
    v2f b = { (kA < Kr) ? Wr[(size_t)kA * 128 + n] : 0.f,
              (kA + 1 < Kr) ? Wr[(size_t)(kA + 1) * 128 + n] : 0.f };
    cr = wmma4(a, b, cr);
  }
  float bb = bl[n];
  for (int r = 0; r < 8; r++) {
    int mr = m0 + r + 8 * hf;
    float inv = 1.f / fmaxf(cnt[mr], 1.f);
    float v = cl[r] * inv + bb + cr[r];
    outp[(size_t)mr * 128 + n] = fmaxf(v, 0.f);
  }
}

// layernorm over 128 channels, one wave per row, in place; grid.x = M/8
__global__ void ln_kernel(float* __restrict__ X, const float* __restrict__ g,
                          const float* __restrict__ b) {
  int row = blockIdx.x * 8 + (threadIdx.x >> 5);
  int lane = threadIdx.x & 31;
  float* p = X + (size_t)row * 128;
  float v0 = p[lane], v1 = p[lane + 32], v2 = p[lane + 64], v3 = p[lane + 96];
  float s = v0 + v1 + v2 + v3;
  for (int o2 = 16; o2; o2 >>= 1) s += __shfl_xor(s, o2, 32);
  float mean = s * (1.f / 128.f);
  float d0 = v0 - mean, d1 = v1 - mean, d2 = v2 - mean, d3 = v3 - mean;
  float q = d0 * d0 + d1 * d1 + d2 * d2 + d3 * d3;
  for (int o2 = 16; o2; o2 >>= 1) q += __shfl_xor(q, o2, 32);
  float inv = rsqrtf(q * (1.f / 128.f) + LNEPS);
  p[lane]      = d0 * inv * g[lane]      + b[lane];
  p[lane + 32] = d1 * inv * g[lane + 32] + b[lane + 32];
  p[lane + 64] = d2 * inv * g[lane + 64] + b[lane + 64];
  p[lane + 96] = d3 * inv * g[lane + 96] + b[lane + 96];
}

// ---------------------------------------------------------------- conv1d as implicit GEMM
// in (32,Cin,Tin), w (64,Cin,8), out (32,64,Tout) with relu. block=128 (4 oc tiles)
__global__ void conv1d_wmma(const float* __restrict__ in, const float* __restrict__ w,
                            const float* __restrict__ bias, float* __restrict__ outp,
                            int Cin, int Tin, int Tout) {
  const int b = blockIdx.y;
  const int t0 = blockIdx.x * 16;
  const int lane = threadIdx.x & 31;
  const int wv = threadIdx.x >> 5;
  const int oc0 = wv * 16;
  const int mrow = lane & 15;
  const int hf = lane >> 4;
  const int tn = t0 + mrow;
  const int tcl = (tn < Tout) ? tn : (Tout - 1);
  const int K = Cin * 8;
  const float* inb = in + (size_t)b * Cin * Tin;
  v8f c = {0.f,0.f,0.f,0.f,0.f,0.f,0.f,0.f};
  for (int k = 0; k < K; k += 4) {
    int kA = k + 2 * hf;
    v2f a = { w[(size_t)(oc0 + mrow) * K + kA],
              w[(size_t)(oc0 + mrow) * K + kA + 1] };
    int g0 = kA >> 3, r0 = kA & 7;
    int g1 = (kA + 1) >> 3, r1 = (kA + 1) & 7;
    v2f bb = { inb[(size_t)g0 * Tin + tcl + r0],
               inb[(size_t)g1 * Tin + tcl + r1] };
    c = wmma4(a, bb, c);
  }
  if (tn < Tout) {
    for (int r = 0; r < 8; r++) {
      int oc = oc0 + r + 8 * hf;
      outp[((size_t)b * 64 + oc) * Tout + tn] = fmaxf(c[r] + bias[oc], 0.f);
    }
  }
}

// ---------------------------------------------------------------- head: d1 epilogue + d2 + bn2 + relu + d3
__global__ void head_kernel(const float* __restrict__ d1acc, const float* __restrict__ d1b,
                            const float* __restrict__ oth, const float* __restrict__ d2W,
                            const float* __restrict__ d2b, const float* __restrict__ bn2g,
                            const float* __restrict__ bn2b, const float* __restrict__ d3W,
                            const float* __restrict__ d3b, float* __restrict__ outp) {
  __shared__ float h[256];
  int b = blockIdx.x, j = threadIdx.x;
  float acc = d2b[j];
  for (int k = 0; k < 128; k++) {
    float ck = fmaxf(d1acc[b * 128 + k] + d1b[k], 0.f);
    acc += ck * d2W[(size_t)k * 256 + j];
  }
  for (int k = 0; k < 128; k++)
    acc += oth[b * 128 + k] * d2W[(size_t)(128 + k) * 256 + j];
  acc = acc * (bn2g[j] * rsqrtf(1.f + LNEPS)) + bn2b[j];
  acc = fmaxf(acc, 0.f);
  h[j] = acc;
  __syncthreads();
  if (j < 2) {
    float o = d3b[j];
    for (int k = 0; k < 256; k++) o += h[k] * d3W[k * 2 + j];
    outp[b * 2 + j] = o;
  }
}

// ================================================================ launch
extern "C" void kernel_launch(void* const* d_in, const int* in_sizes, int n_in,
                              void* d_out, int out_size, void* d_ws, size_t ws_size,
                              hipStream_t stream) {
  const float* x_src   = (const float*)d_in[0];
  const float* x_dst   = (const float*)d_in[1];
  const int*   ei      = (const int*)d_in[2];
  const float* of      = (const float*)d_in[3];
  const float* pnode_W = (const float*)d_in[4];
  const float* pnode_b = (const float*)d_in[5];
  const float* fnode_W = (const float*)d_in[6];
  const float* fnode_b = (const float*)d_in[7];
  const float* cnn1_W  = (const float*)d_in[8];
  const float* cnn1_b  = (const float*)d_in[9];
  const float* cnn2_W  = (const float*)d_in[10];
  const float* cnn2_b  = (const float*)d_in[11];
  const float* bn_g    = (const float*)d_in[12];
  const float* bn_b    = (const float*)d_in[13];
  const float* fcone_W = (const float*)d_in[14];
  const float* fcone_b = (const float*)d_in[15];
  const float* g1l0Wl  = (const float*)d_in[16];
  const float* g1l0b   = (const float*)d_in[17];
  const float* g1l0Wr  = (const float*)d_in[18];
  const float* g2l0Wl  = (const float*)d_in[19];
  const float* g2l0b   = (const float*)d_in[20];
  const float* g2l0Wr  = (const float*)d_in[21];
  const float* g1Wl    = (const float*)d_in[22];
  const float* g1b     = (const float*)d_in[23];
  const float* g1Wr    = (const float*)d_in[24];
  const float* g2Wl    = (const float*)d_in[25];
  const float* g2b     = (const float*)d_in[26];
  const float* g2Wr    = (const float*)d_in[27];
  const float* ln_g    = (const float*)d_in[28];
  const float* ln_b    = (const float*)d_in[29];
  const float* c0_W    = (const float*)d_in[30];
  const float* c0_b    = (const float*)d_in[31];
  const float* c_W     = (const float*)d_in[32];
  const float* c_b     = (const float*)d_in[33];
  const float* d1_W    = (const float*)d_in[34];
  const float* d1_b    = (const float*)d_in[35];
  const float* d2_W    = (const float*)d_in[36];
  const float* d2_b    = (const float*)d_in[37];
  const float* bn2_g   = (const float*)d_in[38];
  const float* bn2_b   = (const float*)d_in[39];
  const float* d3_W    = (const float*)d_in[40];
  const float* d3_b    = (const float*)d_in[41];
  float* outp = (float*)d_out;

  const int E2 = in_sizes[2] / 2;  // 2*E = 2097152
  const int E = E2 / 2;            // 1048576

  float* Wsp = (float*)d_ws;
  size_t o = 0;
  float* xp3  = Wsp + o; o += 393216;     // (131072,3)
  float* xf3  = Wsp + o; o += 6144;       // (2048,3)
  float* cntp = Wsp + o; o += 131072;
  float* cntf = Wsp + o; o += 2048;
  float* aggp = Wsp + o; o += 16777216;   // (131072,128); later reused as y0
  float* aggf = Wsp + o; o += 262144;     // (2048,128)
  float* xpA  = Wsp + o; o += 16777216;   // later reused as y1/y2
  float* xpB  = Wsp + o; o += 16777216;
  float* xfA  = Wsp + o; o += 262144;
  float* xfB  = Wsp + o; o += 262144;
  float* c1o  = Wsp + o; o += 973824;     // (32,32,951)
  float* c2o  = Wsp + o; o += 3694592;    // (32,128,902)
  float* cmax = Wsp + o; o += 4096;
  float* oth  = Wsp + o; o += 4096;
  float* d1a  = Wsp + o; o += 4096;
  float* y0 = aggp;                       // (32,64,4089) = 8374272
  float* y1 = xpA;                        // (32,64,4082) = 8359936
  float* y2 = xpA + 8359936;              // (32,64,4075) = 8345600

  // CNN branch
  conv1_kernel<<<dim3(4, 1024), 256, 0, stream>>>(of, cnn1_W, cnn1_b, c1o);
  conv2_wmma<<<dim3(57, 32), 256, 0, stream>>>(c1o, cnn2_W, cnn2_b, bn_g, bn_b, c2o);
  maxred_kernel<<<512, 256, 0, stream>>>(c2o, cmax);
  fc_kernel<<<16, 256, 0, stream>>>(cmax, fcone_W, fcone_b, oth, 32, 128, 128);

  // node embeddings
  gemm_skinny32<<<dim3(192, 1), 128, 0, stream>>>(x_dst, pnode_W, pnode_b, xp3,
                                                  12288, 12288, 12288, 0);
  fc_kernel<<<24, 256, 0, stream>>>(x_src, fnode_W, fnode_b, xf3, 32, 64, 192);

  // edge degree counts (shared by all layers)
  zero_kernel<<<1024, 256, 0, stream>>>(cntp, 131072 + 2048);
  count_kernel<<<4096, 256, 0, stream>>>(ei, E, E2, 0, 1, cntp);
  count_kernel<<<4096, 256, 0, stream>>>(ei, E, E2, 1, 0, cntf);

  const float* xp_old = xp3;
  const float* xf_old = xf3;
  for (int i = 0; i < 4; i++) {
    float* xp_new = ((i & 1) == 0) ? xpA : xpB;
    float* xf_new = ((i & 1) == 0) ? xfA : xfB;
    int Kp = (i == 0) ? 3 : 128;          // x_f / x_p feature width going in
    const float* Wlp = (i == 0) ? g1l0Wl : g1Wl + (size_t)(i - 1) * 16384;
    const float* blp = (i == 0) ? g1l0b  : g1b  + (size_t)(i - 1) * 128;
    const float* Wrp = (i == 0) ? g1l0Wr : g1Wr + (size_t)(i - 1) * 16384;
    const float* Wlf = (i == 0) ? g2l0Wl : g2Wl + (size_t)(i - 1) * 16384;
    const float* blf = (i == 0) ? g2l0b  : g2b  + (size_t)(i - 1) * 128;
    const float* Wrf = (i == 0) ? g2l0Wr : g2Wr + (size_t)(i - 1) * 16384;

    // p update: aggregate x_f over forward edges
    if (i == 0) {
      zero_kernel<<<1536, 256, 0, stream>>>(aggp, 393216);
      agg3_kernel<<<4096, 256, 0, stream>>>(ei, E, E2, 0, 0, 1, xf_old, aggp);
    } else {
      zero_kernel<<<16384, 256, 0, stream>>>(aggp, 16777216);
      agg128_kernel<<<E, 128, 0, stream>>>(ei, E2, 0, 0, 1, xf_old, aggp);
    }
    sage_fused<<<8192, 256, 0, stream>>>(aggp, cntp, xp_old, Wlp, blp, Wrp,
                                         xp_new, Kp, Kp);
    // f update: aggregate NEW x_p over backward edges
    zero_kernel<<<1024, 256, 0, stream>>>(aggf, 262144);
    agg128_kernel<<<E, 128, 0, stream>>>(ei, E2, 1, 1, 0, xp_new, aggf);
    sage_fused<<<128, 256, 0, stream>>>(aggf, cntf, xf_old, Wlf, blf, Wrf,
                                        xf_new, 128, Kp);
    if (i < 3) {
      ln_kernel<<<16384, 256, 0, stream>>>(xp_new, ln_g + 128 * i, ln_b + 128 * i);
      ln_kernel<<<256, 256, 0, stream>>>(xf_new, ln_g + 128 * i, ln_b + 128 * i);
    }
    xp_old = xp_new;
    xf_old = xf_new;
  }

  // conv1d stack on final x_p (flat (32,128,4096))
  conv1d_wmma<<<dim3(256, 32), 128, 0, stream>>>(xpB, c0_W, c0_b, y0, 128, 4096, 4089);
  conv1d_wmma<<<dim3(256, 32), 128, 0, stream>>>(y0, c_W, c_b, y1, 64, 4089, 4082);
  conv1d_wmma<<<dim3(255, 32), 128, 0, stream>>>(y1, c_W + 32768, c_b + 64, y2, 64, 4082, 4075);

  // d1: (32,260800) @ (260800,128), K-split 64 x 4075 with atomic reduction
  zero_kernel<<<16, 256, 0, stream>>>(d1a, 4096);
  gemm_skinny32<<<dim3(2, 64), 128, 0, stream>>>(y2, d1_W, nullptr, d1a,
                                                 260800, 128, 4075, 1);

  // head: relu(d1)+concat(other) -> d2 -> bn2 -> relu -> d3
  head_kernel<<<32, 256, 0, stream>>>(d1a, d1_b, oth, d2_W, d2_b, bn2_g, bn2_b,
                                      d3_W, d3_b, outp);
}